// DiffusionModel_26628797235675
// MI455X (gfx1250) — compile-verified
//
#include <hip/hip_runtime.h>
#include <cstddef>

// ---------------------------------------------------------------------------
// DiffusionModel GNN for MI455X (gfx1250, wave32).
// Dense GEMMs: V_WMMA_F32_16X16X4_F32, B panel staged transposed in LDS so
// each B fragment is one ds_load_b64.  Edge aggregation: global_atomic_add_f32.
// ---------------------------------------------------------------------------

typedef __attribute__((ext_vector_type(2))) float v2f;
typedef __attribute__((ext_vector_type(8))) float v8f;

#define HDIM 256

__device__ __forceinline__ float silu_f(float y) {
    return y / (1.0f + __expf(-y));
}

// ---------------------------------------------------------------------------
// GEMM: C[M x NC] = act(A[M x K] @ B[K x NC] + bias[NC] + extra[NC])
// Block = 256 threads (8 waves).  Block tile: 256(M) x 64(N).
// Wave tile: 32(M) x 64(N) = 2 m-subtiles x 4 n-subtiles of 16x16.
// B panel is staged in LDS transposed: sBT[n][k], row stride K+2 (keeps
// ds_load_b64 8-byte aligned, kills fill bank conflicts).
// Fragment layouts (ISA 7.12.2, 32-bit operands):
//   A 16x4: lane<16 {A[m][k0],A[m][k0+1]}, lane>=16 {A[m][k0+2],A[m][k0+3]}
//   B 4x16: mirrored (upper lane half carries K+2)   -> one ds_load_b64
//   C/D:    VGPR r -> row 8*half + r, col = lane&15
// ---------------------------------------------------------------------------
template <int ACT>
__global__ __launch_bounds__(256) void gemm_wmma_f32(
    const float* __restrict__ A, const float* __restrict__ B,
    float* __restrict__ C, const float* __restrict__ bias,
    const float* __restrict__ extra, int M, int K, int NC)
{
    extern __shared__ float sBT[];             // [64][K+2]
    const int tid   = threadIdx.x;
    const int wave  = tid >> 5;
    const int lane  = tid & 31;
    const int half  = lane >> 4;               // 0 | 1
    const int l16   = lane & 15;
    const int nbase = blockIdx.x * 64;
    const int ldb   = K + 2;

    // cooperative transpose-fill: sBT[n][k] = B[k][nbase+n]
    for (int idx = tid; idx < 64 * K; idx += 256) {
        const int n = idx & 63;
        const int k = idx >> 6;
        sBT[n * ldb + k] = B[(size_t)k * NC + nbase + n];
    }
    __syncthreads();

    const int m0 = (blockIdx.y * 8 + wave) * 32;   // wave's first row
    if (m0 >= M) return;                           // wave-uniform; no more barriers
    const int r0 = m0 + l16;
    const int r1 = m0 + 16 + l16;
    const float* Arow0 = A + (size_t)(r0 < M ? r0 : M - 1) * K;
    const float* Arow1 = A + (size_t)(r1 < M ? r1 : M - 1) * K;

    v8f acc[2][4] = {};
    for (int k0 = 0; k0 < K; k0 += 4) {
        const int k = k0 + half * 2;
        v2f a[2];
        a[0] = *(const v2f*)(Arow0 + k);
        a[1] = *(const v2f*)(Arow1 + k);
        const float* bp = sBT + l16 * ldb + k;
        v2f b[4];
        b[0] = *(const v2f*)(bp);
        b[1] = *(const v2f*)(bp + 16 * ldb);
        b[2] = *(const v2f*)(bp + 32 * ldb);
        b[3] = *(const v2f*)(bp + 48 * ldb);
#pragma unroll
        for (int i = 0; i < 2; ++i)
#pragma unroll
            for (int j = 0; j < 4; ++j)
                acc[i][j] = __builtin_amdgcn_wmma_f32_16x16x4_f32(
                    false, a[i], false, b[j], (short)0, acc[i][j], false, false);
    }

#pragma unroll
    for (int i = 0; i < 2; ++i) {
        const int mrow = m0 + i * 16 + half * 8;
#pragma unroll
        for (int r = 0; r < 8; ++r) {
            const int row = mrow + r;
            if (row < M) {
                float* Crow = C + (size_t)row * NC + nbase + l16;
#pragma unroll
                for (int j = 0; j < 4; ++j) {
                    const int col = nbase + l16 + j * 16;
                    float y = acc[i][j][r];
                    if (bias)  y += bias[col];
                    if (extra) y += extra[col];
                    if (ACT == 1) y = silu_f(y);
                    Crow[j * 16] = y;
                }
            }
        }
    }
}

// ---------------------------------------------------------------------------
// Time embedding: te = silu(t * w1 + b1) @ w2 + b2     (1 block, 256 threads)
// ---------------------------------------------------------------------------
__global__ __launch_bounds__(HDIM) void time_embed_kernel(
    const float* __restrict__ t, const float* __restrict__ w1,
    const float* __restrict__ b1, const float* __restrict__ w2,
    const float* __restrict__ b2, float* __restrict__ te)
{
    __shared__ float u[HDIM];
    const int j = threadIdx.x;
    u[j] = silu_f(t[0] * w1[j] + b1[j]);
    __syncthreads();
    float acc = b2[j];
    for (int k = 0; k < HDIM; ++k) acc += u[k] * w2[k * HDIM + j];
    te[j] = acc;
}

// ---------------------------------------------------------------------------
// Degree helpers (self-loop -> deg starts at 1, so deg > 0 always)
// ---------------------------------------------------------------------------
__global__ void deg_init_kernel(float* __restrict__ deg, int N) {
    int v = blockIdx.x * 256 + threadIdx.x;
    if (v < N) deg[v] = 1.0f;
}
__global__ void deg_count_kernel(const int* __restrict__ dst, float* __restrict__ deg, int E) {
    int e = blockIdx.x * 256 + threadIdx.x;
    if (e < E) unsafeAtomicAdd(&deg[dst[e]], 1.0f);
}
__global__ void deg_rsqrt_kernel(float* __restrict__ deg, int N) {
    int v = blockIdx.x * 256 + threadIdx.x;
    if (v < N) deg[v] = rsqrtf(deg[v]);
}

// ---------------------------------------------------------------------------
// out[row] = h[row] + gcn_b + xw[row] * dinv[row]^2   (self-loop + bias init)
// grid = N blocks x 256 threads
// ---------------------------------------------------------------------------
__global__ __launch_bounds__(HDIM) void agg_init_kernel(
    const float* __restrict__ h, const float* __restrict__ xw,
    const float* __restrict__ dinv, const float* __restrict__ gcn_b,
    float* __restrict__ out)
{
    const size_t idx = (size_t)blockIdx.x * HDIM + threadIdx.x;
    const float di = dinv[blockIdx.x];
    out[idx] = h[idx] + gcn_b[threadIdx.x] + xw[idx] * di * di;
}

// ---------------------------------------------------------------------------
// Edge scatter: out[dst] += xw[src] * dinv[src]*dinv[dst].  One wave per edge,
// 8 columns per lane, hardware float atomics.
// ---------------------------------------------------------------------------
__global__ __launch_bounds__(256) void scatter_kernel(
    const float* __restrict__ xw, const int* __restrict__ src,
    const int* __restrict__ dst, const float* __restrict__ dinv,
    float* __restrict__ out, int E)
{
    const int e = blockIdx.x * 8 + (threadIdx.x >> 5);
    if (e >= E) return;
    const int lane = threadIdx.x & 31;
    const int s = src[e], d = dst[e];
    const float w = dinv[s] * dinv[d];
    const float* xr = xw + (size_t)s * HDIM;
    float* orow = out + (size_t)d * HDIM;
#pragma unroll
    for (int r = 0; r < 8; ++r) {
        const int j = lane + 32 * r;
        unsafeAtomicAdd(&orow[j], xr[j] * w);
    }
}

// ---------------------------------------------------------------------------
// LayerNorm over 256 cols (eps 1e-5, biased variance), optional SiLU, optional
// accumulate (out += y).  One block (8 waves) per row; shfl + LDS reduction.
// flags: bit0 = silu, bit1 = accumulate into out
// ---------------------------------------------------------------------------
__global__ __launch_bounds__(HDIM) void layernorm_kernel(
    const float* __restrict__ in, float* __restrict__ out,
    const float* __restrict__ g, const float* __restrict__ b, int flags)
{
    __shared__ float s1[8], s2[8], smv[2];
    const int tid = threadIdx.x;
    const size_t base = (size_t)blockIdx.x * HDIM;
    const float x = in[base + tid];
    float v1 = x, v2 = x * x;
#pragma unroll
    for (int off = 16; off > 0; off >>= 1) {
        v1 += __shfl_xor(v1, off, 32);
        v2 += __shfl_xor(v2, off, 32);
    }
    if ((tid & 31) == 0) { s1[tid >> 5] = v1; s2[tid >> 5] = v2; }
    __syncthreads();
    if (tid == 0) {
        float t1 = 0.f, t2 = 0.f;
#pragma unroll
        for (int i = 0; i < 8; ++i) { t1 += s1[i]; t2 += s2[i]; }
        const float mean = t1 * (1.0f / HDIM);
        const float var  = t2 * (1.0f / HDIM) - mean * mean;
        smv[0] = mean;
        smv[1] = rsqrtf(var + 1e-5f);
    }
    __syncthreads();
    float y = (x - smv[0]) * smv[1] * g[tid] + b[tid];
    if (flags & 1) y = silu_f(y);
    if (flags & 2) out[base + tid] += y;
    else           out[base + tid] = y;
}

// ---------------------------------------------------------------------------
extern "C" void kernel_launch(void* const* d_in, const int* in_sizes, int n_in,
                              void* d_out, int out_size, void* d_ws, size_t ws_size,
                              hipStream_t stream)
{
    const float* x        = (const float*)d_in[0];
    const float* t        = (const float*)d_in[1];
    const int*   edge     = (const int*)d_in[2];
    const float* time_w1  = (const float*)d_in[3];
    const float* time_b1  = (const float*)d_in[4];
    const float* time_w2  = (const float*)d_in[5];
    const float* time_b2  = (const float*)d_in[6];
    const float* in_w     = (const float*)d_in[7];
    const float* in_b     = (const float*)d_in[8];
    const float* gcn_w    = (const float*)d_in[9];
    const float* gcn_b    = (const float*)d_in[10];
    const float* res_w1   = (const float*)d_in[11];
    const float* res_b1   = (const float*)d_in[12];
    const float* res_g1   = (const float*)d_in[13];
    const float* res_be1  = (const float*)d_in[14];
    const float* res_w2   = (const float*)d_in[15];
    const float* res_b2   = (const float*)d_in[16];
    const float* res_g2   = (const float*)d_in[17];
    const float* res_be2  = (const float*)d_in[18];
    const float* out_w1   = (const float*)d_in[19];
    const float* out_b1   = (const float*)d_in[20];
    const float* out_w2   = (const float*)d_in[21];
    const float* out_b2   = (const float*)d_in[22];

    const int Dn = 128, H = HDIM, L = 4;
    const int N = in_sizes[0] / Dn;
    const int E = in_sizes[2] / 2;
    const int* src = edge;       // edge_index row 0
    const int* dst = edge + E;   // edge_index row 1

    // workspace carve-out (floats)
    float* ws   = (float*)d_ws;
    float* te   = ws;                               // 256
    float* dinv = ws + 256;                         // N
    const size_t NH = (size_t)N * H;
    float* buf0 = ws + 256 + (((size_t)N + 63) & ~(size_t)63);
    float* buf1 = buf0 + NH;
    float* buf2 = buf1 + NH;

    dim3 blk(256);
    dim3 grd256(H / 64, (N + 255) / 256);           // NC = 256
    dim3 grd128(Dn / 64, (N + 255) / 256);          // NC = 128
    const size_t smemK256 = (size_t)64 * (256 + 2) * sizeof(float);   // 66 KB
    const size_t smemK128 = (size_t)64 * (128 + 2) * sizeof(float);   // 33 KB
    const int nbN = (N + 255) / 256;
    const int nbE = (E + 255) / 256;

    // time embedding + symmetric GCN normalization
    time_embed_kernel<<<1, H, 0, stream>>>(t, time_w1, time_b1, time_w2, time_b2, te);
    deg_init_kernel<<<nbN, 256, 0, stream>>>(dinv, N);
    deg_count_kernel<<<nbE, 256, 0, stream>>>(dst, dinv, E);
    deg_rsqrt_kernel<<<nbN, 256, 0, stream>>>(dinv, N);

    // h = x @ in_w + in_b + te
    gemm_wmma_f32<0><<<grd256, blk, smemK128, stream>>>(x, in_w, buf0, in_b, te, N, Dn, H);

    float* h  = buf0;
    float* sA = buf1;
    float* sB = buf2;
    for (int l = 0; l < L; ++l) {
        const float* gw  = gcn_w  + (size_t)l * H * H;
        const float* gb  = gcn_b  + (size_t)l * H;
        const float* rw1 = res_w1 + (size_t)l * H * H;
        const float* rb1 = res_b1 + (size_t)l * H;
        const float* rg1 = res_g1 + (size_t)l * H;
        const float* re1 = res_be1 + (size_t)l * H;
        const float* rw2 = res_w2 + (size_t)l * H * H;
        const float* rb2 = res_b2 + (size_t)l * H;
        const float* rg2 = res_g2 + (size_t)l * H;
        const float* re2 = res_be2 + (size_t)l * H;

        // xw = h @ gcn_w[l]
        gemm_wmma_f32<0><<<grd256, blk, smemK256, stream>>>(h, gw, sA, nullptr, nullptr, N, H, H);
        // sB = h + gcn_b + self-loop term, then scatter edges into it
        agg_init_kernel<<<N, H, 0, stream>>>(h, sA, dinv, gb, sB);
        scatter_kernel<<<(E + 7) / 8, 256, 0, stream>>>(sA, src, dst, dinv, sB, E);
        // r = silu(LN(sB @ res_w1 + b1))  (in place in sA)
        gemm_wmma_f32<0><<<grd256, blk, smemK256, stream>>>(sB, rw1, sA, rb1, nullptr, N, H, H);
        layernorm_kernel<<<N, H, 0, stream>>>(sA, sA, rg1, re1, /*silu*/1);
        // sB += LN(sA @ res_w2 + b2)   (tmp2 reuses the old-h buffer)
        gemm_wmma_f32<0><<<grd256, blk, smemK256, stream>>>(sA, rw2, h, rb2, nullptr, N, H, H);
        layernorm_kernel<<<N, H, 0, stream>>>(h, sB, rg2, re2, /*accumulate*/2);
        // rotate: new h = sB; old-h buffer becomes scratch
        float* nh = sB; sB = h; h = nh;
    }

    // out = silu(h @ out_w1 + out_b1) @ out_w2 + out_b2
    gemm_wmma_f32<1><<<grd256, blk, smemK256, stream>>>(h, out_w1, sA, out_b1, nullptr, N, H, H);
    gemm_wmma_f32<0><<<grd128, blk, smemK256, stream>>>(sA, out_w2, (float*)d_out, out_b2, nullptr, N, H, Dn);
}